// ConvE_32160715113079
// MI455X (gfx1250) — compile-verified
//
#include <hip/hip_runtime.h>

#define N_ENT   200000
#define N_EDGES 600000
#define N_ETYPES 200
#define D   128
#define NB  8
#define SUB 16

typedef __attribute__((ext_vector_type(2))) float v2f;
typedef __attribute__((ext_vector_type(8))) float v8f;

// ---------------------------------------------------------------------------
// x[i, :] = emb_e[h[i], :]   (float4 vectorized row gather)
// ---------------------------------------------------------------------------
__global__ void __launch_bounds__(256)
gather_rows(const int* __restrict__ h, const float* __restrict__ emb,
            float* __restrict__ x) {
    int tid = blockIdx.x * blockDim.x + threadIdx.x;   // over N_ENT * D/4
    int row = tid >> 5;                                // D/4 == 32
    int c4  = tid & 31;
    if (row >= N_ENT) return;
    const float4* s = (const float4*)(emb + (size_t)h[row] * D);
    float4*       d = (float4*)(x + (size_t)row * D);
    d[c4] = s[c4];
}

// ---------------------------------------------------------------------------
// acc[m, n] = sum_k x[m, k] * w[k, n] + bias[n]
// One wave computes one 16x16 output tile via V_WMMA_F32_16X16X4_F32,
// K-loop of 32 WMMAs. block = (32 lanes, 8 waves): wave y -> column tile y.
// grid.x = N_ENT/16 row strips (exact: 12500). EXEC is all-ones everywhere.
// ---------------------------------------------------------------------------
__global__ void __launch_bounds__(256)
loop_gemm_bias(const float* __restrict__ x, const float* __restrict__ w,
               const float* __restrict__ bias, float* __restrict__ acc) {
    const int lane = threadIdx.x;      // 0..31
    const int half = lane >> 4;        // 0 | 1  (K split for A/B layout)
    const int lidx = lane & 15;        // M index (A) / N index (B,C,D)
    const int n0 = threadIdx.y * 16;   // column tile base
    const int m0 = blockIdx.x  * 16;   // row strip base

    v8f c = {};
    const float* arow = x + (size_t)(m0 + lidx) * D;

    for (int kk = 0; kk < D; kk += 4) {
        const int kb = kk + half * 2;
        // A 16x4 fp32: lanes 0-15 hold K=kk,kk+1 ; lanes 16-31 hold K=kk+2,kk+3
        v2f a;
        a.x = arow[kb];
        a.y = arow[kb + 1];
        // B 4x16 fp32: row K striped across lanes within a VGPR
        v2f b;
        b.x = w[(size_t)kb       * D + n0 + lidx];
        b.y = w[(size_t)(kb + 1) * D + n0 + lidx];
        c = __builtin_amdgcn_wmma_f32_16x16x4_f32(
                /*neg_a=*/false, a, /*neg_b=*/false, b,
                /*c_mod=*/(short)0, c, /*reuse_a=*/false, /*reuse_b=*/false);
    }

    const float bv = bias[n0 + lidx];
    #pragma unroll
    for (int r = 0; r < 8; ++r) {
        // C/D layout: VGPR r -> M = r (lanes 0-15) / M = 8+r (lanes 16-31)
        const int row = m0 + r + half * 8;
        acc[(size_t)row * D + n0 + lidx] = c[r] + bv;
    }
}

// ---------------------------------------------------------------------------
// Per-edge BDD message + scatter:
//   msg[e, b*16+o] = (sum_i x[src[e], b*16+i] * w[rel[e], b, i, o]) * norm[e]
//   acc[dst[e], :] += msg[e, :]        (global_atomic_add_f32)
// One wave per edge; 8 edges per 256-thread block. Source row staged in LDS
// (float4), each lane produces 4 consecutive outputs of its block so the W
// inner loads are aligned float4 and the atomics are lane-coalesced.
// ---------------------------------------------------------------------------
__global__ void __launch_bounds__(256)
edge_msg_scatter(const int* __restrict__ src, const int* __restrict__ dst,
                 const int* __restrict__ rel, const float* __restrict__ norm,
                 const float* __restrict__ x, const float* __restrict__ w,
                 float* __restrict__ acc) {
    __shared__ float sfeat[8][D];
    const int lane = threadIdx.x;   // 0..31
    const int wv   = threadIdx.y;   // 0..7
    const int e    = blockIdx.x * 8 + wv;
    const bool valid = (e < N_EDGES);
    const int ec = valid ? e : (N_EDGES - 1);   // clamp so all waves do safe work

    const int srow = src[ec];
    ((float4*)sfeat[wv])[lane] = ((const float4*)(x + (size_t)srow * D))[lane];
    __syncthreads();

    const int b  = lane >> 2;          // block 0..7
    const int o0 = (lane & 3) * 4;     // first of 4 output cols in block
    const float* wb = w + (size_t)rel[ec] * (NB * SUB * SUB)
                        + (size_t)b * (SUB * SUB) + o0;
    const float* f = &sfeat[wv][b * 16];

    float m0 = 0.f, m1 = 0.f, m2 = 0.f, m3 = 0.f;
    #pragma unroll
    for (int i = 0; i < 16; ++i) {
        const float  fi = f[i];                          // LDS broadcast (4 lanes share)
        const float4 wr = *(const float4*)(wb + i * 16); // aligned 16B, hot in L0/L2
        m0 = fmaf(fi, wr.x, m0);
        m1 = fmaf(fi, wr.y, m1);
        m2 = fmaf(fi, wr.z, m2);
        m3 = fmaf(fi, wr.w, m3);
    }

    if (valid) {
        const float nrm = norm[e];
        float* o = acc + (size_t)dst[e] * D + lane * 4;
        atomicAdd(o + 0, m0 * nrm);
        atomicAdd(o + 1, m1 * nrm);
        atomicAdd(o + 2, m2 * nrm);
        atomicAdd(o + 3, m3 * nrm);
    }
}

// ---------------------------------------------------------------------------
// x = relu(acc)  (float4 vectorized)
// ---------------------------------------------------------------------------
__global__ void __launch_bounds__(256)
relu_copy(const float* __restrict__ acc, float* __restrict__ x, int n4) {
    int tid = blockIdx.x * blockDim.x + threadIdx.x;
    if (tid >= n4) return;
    float4 v = ((const float4*)acc)[tid];
    v.x = fmaxf(v.x, 0.f);
    v.y = fmaxf(v.y, 0.f);
    v.z = fmaxf(v.z, 0.f);
    v.w = fmaxf(v.w, 0.f);
    ((float4*)x)[tid] = v;
}

// ---------------------------------------------------------------------------
extern "C" void kernel_launch(void* const* d_in, const int* in_sizes, int n_in,
                              void* d_out, int out_size, void* d_ws, size_t ws_size,
                              hipStream_t stream) {
    const int*   h    = (const int*)  d_in[0];
    const int*   src  = (const int*)  d_in[1];
    const int*   dst  = (const int*)  d_in[2];
    const int*   rel  = (const int*)  d_in[3];
    const float* norm = (const float*)d_in[4];
    const float* emb  = (const float*)d_in[5];
    const float* w1   = (const float*)d_in[6];
    const float* lw1  = (const float*)d_in[7];
    const float* b1   = (const float*)d_in[8];
    const float* w2   = (const float*)d_in[9];
    const float* lw2  = (const float*)d_in[10];
    const float* b2   = (const float*)d_in[11];

    float* out = (float*)d_out;        // N_ENT x D, doubles as layer accumulator
    float* x   = (float*)d_ws;         // N_ENT x D fp32 feature buffer (102.4 MB)

    const int n4 = N_ENT * D / 4;
    const dim3 waveBlk(32, 8);

    // x = emb_e[h]
    gather_rows<<<(n4 + 255) / 256, 256, 0, stream>>>(h, emb, x);

    // ---- layer 1 (acc lives in d_out) ----
    loop_gemm_bias<<<N_ENT / 16, waveBlk, 0, stream>>>(x, lw1, b1, out);
    edge_msg_scatter<<<(N_EDGES + 7) / 8, waveBlk, 0, stream>>>(src, dst, rel, norm,
                                                                x, w1, out);
    relu_copy<<<(n4 + 255) / 256, 256, 0, stream>>>(out, x, n4);

    // ---- layer 2 (final result directly in d_out, no relu) ----
    loop_gemm_bias<<<N_ENT / 16, waveBlk, 0, stream>>>(x, lw2, b2, out);
    edge_msg_scatter<<<(N_EDGES + 7) / 8, waveBlk, 0, stream>>>(src, dst, rel, norm,
                                                                x, w2, out);
}